// MLP_VQVAE_54941221651335
// MI455X (gfx1250) — compile-verified
//
#include <hip/hip_runtime.h>
#include <math.h>

typedef float v2f __attribute__((ext_vector_type(2)));
typedef float v8f __attribute__((ext_vector_type(8)));

#define BDIM 16384
#define LAT 256
#define KCODES 8192

__device__ __forceinline__ float gelu_exact(float x) {
    return 0.5f * x * (1.0f + erff(x * 0.70710678118654752440f));
}

// ---------------------------------------------------------------------------
// Generic f32 WMMA GEMM: C[M,N] = act(A[M,K] @ W[K,N] + bias[N])
// Block tile 128(M) x 64(N), Kt = 32. 128 threads = 4 waves, each wave a
// 64x32 sub-tile = 4x2 grid of 16x16 WMMA tiles, K stepped by 4 with
// V_WMMA_F32_16X16X4_F32.
// ---------------------------------------------------------------------------
template <bool GELU>
__global__ __launch_bounds__(128) void gemm_bias_act(
    const float* __restrict__ A, const float* __restrict__ W,
    const float* __restrict__ bias, float* __restrict__ C,
    int M, int N, int K)
{
    __shared__ float As[128][36];  // [m][k], stride 36 floats (16B multiple)
    __shared__ float Bs[64][36];   // [n][k] (transposed during staging)

    const int tid  = threadIdx.x;
    const int lane = tid & 31;
    const int wave = tid >> 5;
    const int wm   = (wave & 1) * 64;
    const int wn   = (wave >> 1) * 32;
    const int l16  = lane & 15;
    const int half = lane >> 4;          // 0: K=4s,4s+1 ; 1: K=4s+2,4s+3
    const int blockM = blockIdx.y * 128;
    const int blockN = blockIdx.x * 64;

    v8f acc[4][2];
    v8f vzero = {};
#pragma unroll
    for (int mi = 0; mi < 4; ++mi)
#pragma unroll
        for (int ni = 0; ni < 2; ++ni) acc[mi][ni] = vzero;

    for (int k0 = 0; k0 < K; k0 += 32) {
        // Stage A tile: 128 rows x 32 cols = 1024 float4 slots
#pragma unroll
        for (int j = 0; j < 8; ++j) {
            int slot = tid + 128 * j;
            int row  = slot >> 3;
            int c4   = (slot & 7) << 2;
            float4 v = *(const float4*)(A + (size_t)(blockM + row) * K + k0 + c4);
            *(float4*)(&As[row][c4]) = v;
        }
        // Stage B tile transposed: W[k0+kr][blockN+n..] -> Bs[n][kr]
#pragma unroll
        for (int j = 0; j < 4; ++j) {
            int slot = tid + 128 * j;           // 512 float4 slots
            int kr   = slot >> 4;
            int n4   = (slot & 15) << 2;
            float4 v = *(const float4*)(W + (size_t)(k0 + kr) * N + blockN + n4);
            Bs[n4 + 0][kr] = v.x;
            Bs[n4 + 1][kr] = v.y;
            Bs[n4 + 2][kr] = v.z;
            Bs[n4 + 3][kr] = v.w;
        }
        __syncthreads();

#pragma unroll
        for (int s = 0; s < 8; ++s) {
            const int kk = 4 * s + 2 * half;
            v2f af[4], bf[2];
#pragma unroll
            for (int mi = 0; mi < 4; ++mi)
                af[mi] = *(const v2f*)(&As[wm + 16 * mi + l16][kk]);
#pragma unroll
            for (int ni = 0; ni < 2; ++ni)
                bf[ni] = *(const v2f*)(&Bs[wn + 16 * ni + l16][kk]);
#pragma unroll
            for (int mi = 0; mi < 4; ++mi)
#pragma unroll
                for (int ni = 0; ni < 2; ++ni)
                    acc[mi][ni] = __builtin_amdgcn_wmma_f32_16x16x4_f32(
                        false, af[mi], false, bf[ni], (short)0, acc[mi][ni],
                        false, false);
        }
        __syncthreads();
    }

    // Epilogue: C layout — VGPR r: lanes 0-15 M=r, lanes 16-31 M=r+8, N=l16
#pragma unroll
    for (int mi = 0; mi < 4; ++mi) {
#pragma unroll
        for (int ni = 0; ni < 2; ++ni) {
            int col  = blockN + wn + 16 * ni + l16;
            float bv = bias[col];
#pragma unroll
            for (int r = 0; r < 8; ++r) {
                int row = blockM + wm + 16 * mi + r + 8 * half;
                float v = acc[mi][ni][r] + bv;
                if (GELU) v = gelu_exact(v);
                C[(size_t)row * N + col] = v;
            }
        }
    }
}

// ---------------------------------------------------------------------------
// VQ argmin over 8192 codes: dist ~ ||e||^2 - 2 z.e  (||z||^2 row-constant).
// 64 threads = 2 waves; block owns 128 rows (z_e resident in LDS), each wave
// a 64x64 WMMA tile per 64-code chunk; running per-lane (min, idx) then a
// 16-lane shuffle reduction with first-index tie-breaking.
// ---------------------------------------------------------------------------
__global__ __launch_bounds__(64) void vq_argmin(
    const float* __restrict__ Ze, const float* __restrict__ Emb,
    const float* __restrict__ Enorm, int* __restrict__ IdxInt,
    float* __restrict__ IdxF)
{
    extern __shared__ float smem[];
    float* Zs = smem;                 // [128][260]
    float* Es = smem + 128 * 260;     // [64][68]

    const int tid  = threadIdx.x;     // 0..63
    const int lane = tid & 31;
    const int wave = tid >> 5;
    const int wm   = wave * 64;
    const int l16  = lane & 15;
    const int half = lane >> 4;
    const int blockM = blockIdx.x * 128;

    // Stage z_e tile: 128 x 256 = 8192 float4 slots, 64 threads
    for (int j = 0; j < 128; ++j) {
        int slot = tid + 64 * j;
        int row  = slot >> 6;
        int c4   = (slot & 63) << 2;
        *(float4*)(&Zs[row * 260 + c4]) =
            *(const float4*)(Ze + (size_t)(blockM + row) * LAT + c4);
    }
    __syncthreads();

    float minv[4][8];
    int   mini[4][8];
#pragma unroll
    for (int mi = 0; mi < 4; ++mi)
#pragma unroll
        for (int r = 0; r < 8; ++r) { minv[mi][r] = 3.402823466e38f; mini[mi][r] = 0; }

    for (int n0 = 0; n0 < KCODES; n0 += 64) {
        v8f acc[4][4];
        v8f vzero = {};
#pragma unroll
        for (int mi = 0; mi < 4; ++mi)
#pragma unroll
            for (int ni = 0; ni < 4; ++ni) acc[mi][ni] = vzero;

        for (int kt = 0; kt < LAT; kt += 64) {
            // Stage codebook tile Es[n][k] = Emb[n0+n][kt+k] (64x64)
#pragma unroll
            for (int j = 0; j < 16; ++j) {
                int slot = tid + 64 * j;
                int row  = slot >> 4;
                int c4   = (slot & 15) << 2;
                *(float4*)(&Es[row * 68 + c4]) =
                    *(const float4*)(Emb + (size_t)(n0 + row) * LAT + kt + c4);
            }
            __syncthreads();
#pragma unroll
            for (int s = 0; s < 16; ++s) {
                const int kk = 4 * s + 2 * half;
                v2f af[4], bf[4];
#pragma unroll
                for (int mi = 0; mi < 4; ++mi)
                    af[mi] = *(const v2f*)(&Zs[(wm + 16 * mi + l16) * 260 + kt + kk]);
#pragma unroll
                for (int ni = 0; ni < 4; ++ni)
                    bf[ni] = *(const v2f*)(&Es[(16 * ni + l16) * 68 + kk]);
#pragma unroll
                for (int mi = 0; mi < 4; ++mi)
#pragma unroll
                    for (int ni = 0; ni < 4; ++ni)
                        acc[mi][ni] = __builtin_amdgcn_wmma_f32_16x16x4_f32(
                            false, af[mi], false, bf[ni], (short)0, acc[mi][ni],
                            false, false);
            }
            __syncthreads();
        }
        // Compare against running minimum
#pragma unroll
        for (int ni = 0; ni < 4; ++ni) {
            int col  = n0 + 16 * ni + l16;
            float en = Enorm[col];
#pragma unroll
            for (int mi = 0; mi < 4; ++mi)
#pragma unroll
                for (int r = 0; r < 8; ++r) {
                    float d = en - 2.0f * acc[mi][ni][r];
                    if (d < minv[mi][r]) { minv[mi][r] = d; mini[mi][r] = col; }
                }
        }
    }

    // Reduce across the 16 lanes holding the same row (xor masks stay in-half)
#pragma unroll
    for (int off = 8; off >= 1; off >>= 1) {
#pragma unroll
        for (int mi = 0; mi < 4; ++mi)
#pragma unroll
            for (int r = 0; r < 8; ++r) {
                float ov = __shfl_xor(minv[mi][r], off);
                int   oi = __shfl_xor(mini[mi][r], off);
                if (ov < minv[mi][r] ||
                    (ov == minv[mi][r] && oi < mini[mi][r])) {
                    minv[mi][r] = ov; mini[mi][r] = oi;
                }
            }
    }
    if (l16 == 0) {
#pragma unroll
        for (int mi = 0; mi < 4; ++mi)
#pragma unroll
            for (int r = 0; r < 8; ++r) {
                int row = blockM + wm + 16 * mi + r + 8 * half;
                IdxInt[row] = mini[mi][r];
                IdxF[row]   = (float)mini[mi][r];
            }
    }
}

// ---------------------------------------------------------------------------
// Small helper / elementwise kernels
// ---------------------------------------------------------------------------
__device__ __forceinline__ float block_sum256(float v) {
    __shared__ float red[8];
    for (int off = 16; off; off >>= 1) v += __shfl_xor(v, off);
    int lane = threadIdx.x & 31, w = threadIdx.x >> 5;
    if (lane == 0) red[w] = v;
    __syncthreads();
    if (threadIdx.x == 0) {
        float s = 0.f;
        for (int j = 0; j < 8; ++j) s += red[j];
        red[0] = s;
    }
    __syncthreads();
    float out = red[0];
    __syncthreads();
    return out;
}

__global__ __launch_bounds__(256) void layernorm_inplace(
    float* __restrict__ X, const float* __restrict__ g,
    const float* __restrict__ b, int N)
{
    size_t base = (size_t)blockIdx.x * N;
    float s = 0.f;
    for (int i = threadIdx.x; i < N; i += 256) s += X[base + i];
    float mean = block_sum256(s) / (float)N;
    float s2 = 0.f;
    for (int i = threadIdx.x; i < N; i += 256) {
        float d = X[base + i] - mean;
        s2 += d * d;
    }
    float var = block_sum256(s2) / (float)N;
    float inv = rsqrtf(var + 1e-5f);
    for (int i = threadIdx.x; i < N; i += 256)
        X[base + i] = (X[base + i] - mean) * inv * g[i] + b[i];
}

__global__ __launch_bounds__(256) void pack_x0(
    const float* __restrict__ obs, const float* __restrict__ traj,
    float* __restrict__ X0)
{
    size_t i = (size_t)blockIdx.x * 256 + threadIdx.x;  // 16384*960 elems
    int row = (int)(i / 960);
    int c   = (int)(i % 960);
    X0[i] = (c < 640) ? obs[(size_t)row * 640 + c]
                      : traj[(size_t)row * 320 + (c - 640)];
}

__global__ __launch_bounds__(32) void enorm_kernel(
    const float* __restrict__ emb, float* __restrict__ enorm)
{
    int k = blockIdx.x, lane = threadIdx.x;
    const float* e = emb + (size_t)k * LAT;
    float s = 0.f;
    for (int j = lane; j < LAT; j += 32) { float v = e[j]; s += v * v; }
    for (int off = 16; off; off >>= 1) s += __shfl_xor(s, off);
    if (lane == 0) enorm[k] = s;
}

__global__ __launch_bounds__(256) void gather_pack(
    const float* __restrict__ obs, const float* __restrict__ emb,
    const int* __restrict__ idx, float* __restrict__ zq_out,
    float* __restrict__ Xd)
{
    int row = blockIdx.x, t = threadIdx.x;
    int k = idx[row];
    float zv = emb[(size_t)k * LAT + t];
    zq_out[(size_t)row * LAT + t] = zv;
    Xd[(size_t)row * 896 + 640 + t] = zv;
    for (int c = t; c < 640; c += 256)
        Xd[(size_t)row * 896 + c] = obs[(size_t)row * 640 + c];
}

__global__ __launch_bounds__(32) void vq_scatter(
    const float* __restrict__ Ze, const int* __restrict__ idx,
    float* __restrict__ counts, float* __restrict__ dw)
{
    int row = blockIdx.x, lane = threadIdx.x;
    int k = idx[row];
    if (lane == 0) atomicAdd(&counts[k], 1.0f);
    const float* z = Ze + (size_t)row * LAT;
    float* d = dw + (size_t)k * LAT;
    for (int j = lane; j < LAT; j += 32) atomicAdd(&d[j], z[j]);
}

__global__ __launch_bounds__(256) void ema_count_kernel(
    const float* __restrict__ ec, const float* __restrict__ counts,
    float* __restrict__ nc)
{
    int i = blockIdx.x * 256 + threadIdx.x;  // 8192
    nc[i] = ec[i] * 0.99f + 0.01f * counts[i];
}

__global__ __launch_bounds__(256) void ema_total_kernel(
    const float* __restrict__ nc, float* __restrict__ total)
{
    float s = 0.f;
    for (int i = threadIdx.x; i < KCODES; i += 256) s += nc[i];
    float t = block_sum256(s);
    if (threadIdx.x == 0) *total = t;
}

__global__ __launch_bounds__(256) void ema_final_kernel(
    const float* __restrict__ nc, const float* __restrict__ total,
    const float* __restrict__ ema_weight, const float* __restrict__ dw,
    float* __restrict__ new_count, float* __restrict__ new_weight,
    float* __restrict__ new_embedding)
{
    int tid = blockIdx.x * 256 + threadIdx.x;  // 8192*256
    int k = tid >> 8;
    float t = *total;
    float ncn = (nc[k] + 1e-5f) / (t + (float)KCODES * 1e-5f) * t;
    float nw = ema_weight[tid] * 0.99f + 0.01f * dw[tid];
    new_weight[tid] = nw;
    new_embedding[tid] = nw / ncn;
    if ((tid & 255) == 0) new_count[k] = ncn;
}

// ---------------------------------------------------------------------------
extern "C" void kernel_launch(void* const* d_in, const int* in_sizes, int n_in,
                              void* d_out, int out_size, void* d_ws, size_t ws_size,
                              hipStream_t stream)
{
    (void)in_sizes; (void)n_in; (void)out_size; (void)ws_size;

    const float* obs  = (const float*)d_in[0];
    const float* traj = (const float*)d_in[1];
    const float* eW0 = (const float*)d_in[2];
    const float* eb0 = (const float*)d_in[3];
    const float* eW1 = (const float*)d_in[4];
    const float* eb1 = (const float*)d_in[5];
    const float* eg1 = (const float*)d_in[6];
    const float* ebt1= (const float*)d_in[7];
    const float* eW2 = (const float*)d_in[8];
    const float* eb2 = (const float*)d_in[9];
    const float* eg2 = (const float*)d_in[10];
    const float* ebt2= (const float*)d_in[11];
    const float* dW0 = (const float*)d_in[12];
    const float* db0 = (const float*)d_in[13];
    const float* dW1 = (const float*)d_in[14];
    const float* db1 = (const float*)d_in[15];
    const float* dg1 = (const float*)d_in[16];
    const float* dbt1= (const float*)d_in[17];
    const float* dW2 = (const float*)d_in[18];
    const float* db2 = (const float*)d_in[19];
    const float* dg2 = (const float*)d_in[20];
    const float* dbt2= (const float*)d_in[21];
    const float* emb       = (const float*)d_in[22];
    const float* ema_count = (const float*)d_in[23];
    const float* ema_wt    = (const float*)d_in[24];

    // Output regions (all float)
    float* out   = (float*)d_out;                      // [16384, 960]
    float* zq    = out   + (size_t)BDIM * 960;         // [16384, 256]
    float* ze    = zq    + (size_t)BDIM * LAT;         // [16384, 256]
    float* idxf  = ze    + (size_t)BDIM * LAT;         // [16384]
    float* nemb  = idxf  + BDIM;                       // [8192, 256]
    float* ncnt  = nemb  + (size_t)KCODES * LAT;       // [8192]
    float* nwt   = ncnt  + KCODES;                     // [8192, 256]

    // Workspace layout (floats)
    float* ws    = (float*)d_ws;
    float* X0    = ws;                                 // [16384,960] -> later Xd [16384,896]
    float* H1    = X0 + (size_t)BDIM * 960;            // [16384,2048] -> later Hd2
    float* H2    = H1 + (size_t)BDIM * 2048;           // [16384,1024] -> later Hd1
    float* enrm  = H2 + (size_t)BDIM * 1024;           // [8192]
    int*   idxi  = (int*)(enrm + KCODES);              // [16384]
    float* cnts  = (float*)(idxi + BDIM);              // [8192]
    float* dw    = cnts + KCODES;                      // [8192,256]
    float* nc    = dw + (size_t)KCODES * LAT;          // [8192]
    float* total = nc + KCODES;                        // [1]

    // ---- Encoder ----
    pack_x0<<<(BDIM * 960) / 256, 256, 0, stream>>>(obs, traj, X0);

    gemm_bias_act<true><<<dim3(2048 / 64, BDIM / 128), 128, 0, stream>>>(
        X0, eW0, eb0, H1, BDIM, 2048, 960);
    layernorm_inplace<<<BDIM, 256, 0, stream>>>(H1, eg1, ebt1, 2048);
    gemm_bias_act<true><<<dim3(1024 / 64, BDIM / 128), 128, 0, stream>>>(
        H1, eW1, eb1, H2, BDIM, 1024, 2048);
    layernorm_inplace<<<BDIM, 256, 0, stream>>>(H2, eg2, ebt2, 1024);
    gemm_bias_act<true><<<dim3(LAT / 64, BDIM / 128), 128, 0, stream>>>(
        H2, eW2, eb2, ze, BDIM, LAT, 1024);

    // ---- Vector quantization ----
    enorm_kernel<<<KCODES, 32, 0, stream>>>(emb, enrm);
    size_t argmin_lds = (size_t)(128 * 260 + 64 * 68) * sizeof(float);
    vq_argmin<<<BDIM / 128, 64, argmin_lds, stream>>>(ze, emb, enrm, idxi, idxf);
    gather_pack<<<BDIM, 256, 0, stream>>>(obs, emb, idxi, zq, X0 /*Xd*/);

    // ---- EMA codebook update ----
    hipMemsetAsync(cnts, 0, (size_t)(KCODES + KCODES * LAT) * sizeof(float), stream);
    vq_scatter<<<BDIM, 32, 0, stream>>>(ze, idxi, cnts, dw);
    ema_count_kernel<<<KCODES / 256, 256, 0, stream>>>(ema_count, cnts, nc);
    ema_total_kernel<<<1, 256, 0, stream>>>(nc, total);
    ema_final_kernel<<<(KCODES * LAT) / 256, 256, 0, stream>>>(
        nc, total, ema_wt, dw, ncnt, nwt, nemb);

    // ---- Decoder ----
    gemm_bias_act<true><<<dim3(1024 / 64, BDIM / 128), 128, 0, stream>>>(
        X0 /*Xd*/, dW0, db0, H2 /*Hd1*/, BDIM, 1024, 896);
    layernorm_inplace<<<BDIM, 256, 0, stream>>>(H2, dg1, dbt1, 1024);
    gemm_bias_act<true><<<dim3(2048 / 64, BDIM / 128), 128, 0, stream>>>(
        H2, dW1, db1, H1 /*Hd2*/, BDIM, 2048, 1024);
    layernorm_inplace<<<BDIM, 256, 0, stream>>>(H1, dg2, dbt2, 2048);
    gemm_bias_act<false><<<dim3(960 / 64, BDIM / 128), 128, 0, stream>>>(
        H1, dW2, db2, out, BDIM, 960, 2048);
}